// MultiHeadAttentionQuantum_65481071396433
// MI455X (gfx1250) — compile-verified
//
#include <hip/hip_runtime.h>

#define B_    4
#define S_    512
#define E_    64
#define H_    8
#define DK_   8
#define DEPTH_ 2
#define NTOK  (B_ * S_ * H_)   // 16384 tokens, one wave32 each

typedef __attribute__((ext_vector_type(2))) float v2f;
typedef __attribute__((ext_vector_type(8))) float v8f;

// ---------------- 8-qubit statevector gates ------------------------------
// amp index a = lane + 32*j ; wire w <-> bit p = 7-w of a.
// bits 0..4 : lane bits, bits 5..7 : register-index bits.

__device__ __forceinline__ void apply_rx(float re[8], float im[8],
                                         float c, float s, int p, int lane) {
  if (p >= 5) {                       // pair lives across register index
    const int r = p - 5;
    #pragma unroll
    for (int j = 0; j < 8; ++j) {
      if (((j >> r) & 1) == 0) {
        const int k = j | (1 << r);
        float r0 = re[j], i0 = im[j], r1 = re[k], i1 = im[k];
        re[j] = c * r0 + s * i1;  im[j] = c * i0 - s * r1;
        re[k] = c * r1 + s * i0;  im[k] = c * i1 - s * r0;
      }
    }
  } else {                            // pair lives across lanes
    const int mask = 1 << p;
    #pragma unroll
    for (int j = 0; j < 8; ++j) {
      float ore = __shfl_xor(re[j], mask, 32);
      float oim = __shfl_xor(im[j], mask, 32);
      float nr = c * re[j] + s * oim;   // symmetric for both halves of pair
      float ni = c * im[j] - s * ore;
      re[j] = nr; im[j] = ni;
    }
  }
}

__device__ __forceinline__ void apply_rz(float re[8], float im[8],
                                         float c, float s, int p, int lane) {
  // diag(e^{-i t/2}, e^{+i t/2}) : multiply by (c + i*sg*s), sg = bit? +s : -s
  #pragma unroll
  for (int j = 0; j < 8; ++j) {
    const int bit = (p >= 5) ? ((j >> (p - 5)) & 1) : ((lane >> p) & 1);
    const float sg = bit ? s : -s;
    float nr = c * re[j] - sg * im[j];
    float ni = c * im[j] + sg * re[j];
    re[j] = nr; im[j] = ni;
  }
}

__device__ __forceinline__ void apply_cnot(float re[8], float im[8],
                                           int pc, int pt, int lane) {
  if (pc >= 5 && pt >= 5) {                 // ctrl+tgt both register bits
    const int rc = pc - 5, rt = pt - 5;
    #pragma unroll
    for (int j = 0; j < 8; ++j) {
      if (((j >> rc) & 1) && !((j >> rt) & 1)) {
        const int k = j | (1 << rt);
        float tr = re[j], ti = im[j];
        re[j] = re[k]; im[j] = im[k];
        re[k] = tr;    im[k] = ti;
      }
    }
  } else if (pc >= 5) {                     // ctrl register bit, tgt lane bit
    const int rc = pc - 5, mask = 1 << pt;
    #pragma unroll
    for (int j = 0; j < 8; ++j) {
      float ore = __shfl_xor(re[j], mask, 32);
      float oim = __shfl_xor(im[j], mask, 32);
      if ((j >> rc) & 1) { re[j] = ore; im[j] = oim; }
    }
  } else if (pt >= 5) {                     // ctrl lane bit, tgt register bit
    const int rt = pt - 5;
    const bool sel = ((lane >> pc) & 1) != 0;
    #pragma unroll
    for (int j = 0; j < 8; ++j) {
      if (((j >> rt) & 1) == 0) {
        const int k = j | (1 << rt);
        float tr = re[j], ti = im[j];
        re[j] = sel ? re[k] : re[j];  im[j] = sel ? im[k] : im[j];
        re[k] = sel ? tr    : re[k];  im[k] = sel ? ti    : im[k];
      }
    }
  } else {                                  // ctrl+tgt both lane bits
    const int mask = 1 << pt;
    const bool sel = ((lane >> pc) & 1) != 0;
    #pragma unroll
    for (int j = 0; j < 8; ++j) {
      float ore = __shfl_xor(re[j], mask, 32);
      float oim = __shfl_xor(im[j], mask, 32);
      re[j] = sel ? ore : re[j];
      im[j] = sel ? oim : im[j];
    }
  }
}

// ---------------- kernel 1: quantum layer, one wave per token ------------
__global__ __launch_bounds__(256) void qsim_kernel(
    const float* __restrict__ x, const float* __restrict__ prx,
    const float* __restrict__ prz, float* __restrict__ zout) {
  const int lane = threadIdx.x & 31;
  const int t = blockIdx.x * 8 + (threadIdx.x >> 5);   // token = (b*S+s)*H + h
  const float* xin = x + (size_t)t * DK_;              // x[b,s, h*8 .. h*8+7]

  float re[8], im[8];
  #pragma unroll
  for (int j = 0; j < 8; ++j) { re[j] = 0.f; im[j] = 0.f; }
  re[0] = (lane == 0) ? 1.f : 0.f;                     // |00000000>

  // data-encoding RX (per-token angles)
  #pragma unroll
  for (int w = 0; w < 8; ++w) {
    const float th = 0.5f * xin[w];
    apply_rx(re, im, __cosf(th), __sinf(th), 7 - w, lane);
  }
  // parameterized layers
  #pragma unroll
  for (int l = 0; l < DEPTH_; ++l) {
    #pragma unroll
    for (int w = 0; w < 8; ++w) {
      const float ax = 0.5f * prx[l * 8 + w];
      apply_rx(re, im, __cosf(ax), __sinf(ax), 7 - w, lane);
      const float az = 0.5f * prz[l * 8 + w];
      apply_rz(re, im, __cosf(az), __sinf(az), 7 - w, lane);
    }
    #pragma unroll
    for (int w = 0; w < 7; ++w)
      apply_cnot(re, im, 7 - w, 6 - w, lane);          // CNOT(w, w+1)
    apply_cnot(re, im, 0, 7, lane);                    // CNOT(7, 0)
  }

  // PauliZ expectations
  float prob[8], psum = 0.f;
  #pragma unroll
  for (int j = 0; j < 8; ++j) { prob[j] = re[j] * re[j] + im[j] * im[j]; psum += prob[j]; }

  float ex[8];
  #pragma unroll
  for (int w = 0; w < 8; ++w) {
    const int p = 7 - w;
    if (p >= 5) {
      const int r = p - 5; float v = 0.f;
      #pragma unroll
      for (int j = 0; j < 8; ++j) v += ((j >> r) & 1) ? -prob[j] : prob[j];
      ex[w] = v;
    } else {
      ex[w] = ((lane >> p) & 1) ? -psum : psum;
    }
  }
  #pragma unroll
  for (int off = 16; off >= 1; off >>= 1) {
    #pragma unroll
    for (int w = 0; w < 8; ++w) ex[w] += __shfl_xor(ex[w], off, 32);
  }
  if (lane == 0) {
    float* o = zout + (size_t)t * DK_;                 // same offset as input
    #pragma unroll
    for (int w = 0; w < 8; ++w) o[w] = ex[w];
  }
}

// ---------------- kernel 2: Y = Z @ W^T via V_WMMA_F32_16X16X4_F32 -------
// In-place on d_out: each wave owns a 16-row M-tile (reads all its rows,
// computes all four 16-col N-tiles, then stores) -> no cross-wave hazard.
__global__ __launch_bounds__(128) void combine_wmma(
    const float* __restrict__ Wc, float* Y) {
  const int lane = threadIdx.x & 31;
  const int mt = blockIdx.x * 4 + (threadIdx.x >> 5);  // 128 M-tiles total
  const int m0 = mt * 16;
  const int half = lane >> 4;        // K pair select: 0 -> K{0,1}, 1 -> K{2,3}
  const int l16 = lane & 15;         // A: row M ; B: col N

  const float* zr = Y + (size_t)(m0 + l16) * E_;
  v8f acc[4] = {};

  #pragma unroll
  for (int kk = 0; kk < 16; ++kk) {  // K = 64 in steps of 4
    const int k0 = kk * 4 + half * 2;
    v2f a; a.x = zr[k0]; a.y = zr[k0 + 1];             // A[m, k] fragment
    #pragma unroll
    for (int n = 0; n < 4; ++n) {
      const float* wr = Wc + (size_t)(n * 16 + l16) * E_;
      v2f b; b.x = wr[k0]; b.y = wr[k0 + 1];           // B[k, e] = W[e, k]
      acc[n] = __builtin_amdgcn_wmma_f32_16x16x4_f32(
          false, a, false, b, (short)0, acc[n], false, false);
    }
  }
  #pragma unroll
  for (int n = 0; n < 4; ++n) {
    #pragma unroll
    for (int r = 0; r < 8; ++r) {    // C/D layout: VGPR r -> M = r + 8*half
      Y[(size_t)(m0 + half * 8 + r) * E_ + n * 16 + l16] = acc[n][r];
    }
  }
}

// ---------------- launch -------------------------------------------------
extern "C" void kernel_launch(void* const* d_in, const int* in_sizes, int n_in,
                              void* d_out, int out_size, void* d_ws, size_t ws_size,
                              hipStream_t stream) {
  const float* x   = (const float*)d_in[0];
  const float* prx = (const float*)d_in[1];
  const float* prz = (const float*)d_in[2];
  const float* Wc  = (const float*)d_in[3];
  float* Y = (float*)d_out;

  qsim_kernel<<<NTOK / 8, 256, 0, stream>>>(x, prx, prz, Y);
  combine_wmma<<<(B_ * S_ / 16) / 4, 128, 0, stream>>>(Wc, Y);
}